// StraightThroughSubsetSampler_32598801776714
// MI455X (gfx1250) — compile-verified
//
#include <hip/hip_runtime.h>
#include <stdint.h>

// Straight-through Gumbel top-k sampler, forward value == k-hot mask.
// Per row (8192 f32): g = scores/tau + noise; top-32 -> 1.0, else 0.0.
//
// MI455X plan: memory-bound (768MB @ 23.3TB/s ~ 33us). CDNA5 async
// global->LDS copies (ASYNCcnt) stage each row; exact MSB-first radix
// select (4x 8-bit passes, LDS histogram + parallel suffix-scan digit
// pick) finds the 32nd-largest key; ties broken by smallest index to
// match jax.lax.top_k stability.

#define N_NODES 8192
#define BLOCK 256
#define VEC4_PER_THREAD (N_NODES / BLOCK / 4)   // 8 float4 groups per thread

// dynamic LDS layout (bytes):
//  [0,      32768) bufS : row scores, later reused as sortable keys (uint)
//  [32768,  65536) bufN : row gumbel noise
//  [65536,  66560) hist : 256 x u32 radix histogram
//  [66560,  67584) scan : 256 x u32 suffix-scan workspace
//  [67584,  71680) eqlist : up to 1024 tie indices
//  [71680,  71744) shv  : scalars {pref, kneed, eqcnt, selcnt}
#define SMEM_BYTES 71744

__device__ __forceinline__ uint32_t key_of(float f) {
  uint32_t u = __float_as_uint(f);
  // map float -> uint with same total order (descending handled by >)
  return (u & 0x80000000u) ? ~u : (u | 0x80000000u);
}

__global__ __launch_bounds__(BLOCK) void topk_khot_kernel(
    const float* __restrict__ scores,
    const float* __restrict__ noise,
    const float* __restrict__ tau_p,
    const int* __restrict__ k_p,
    float* __restrict__ out)
{
  extern __shared__ unsigned char smem[];
  float*     bufS   = (float*)smem;
  float*     bufN   = (float*)(smem + 32768);
  uint32_t*  keys   = (uint32_t*)smem;               // aliases bufS (after use)
  uint32_t*  hist   = (uint32_t*)(smem + 65536);
  uint32_t*  scan   = (uint32_t*)(smem + 66560);
  uint32_t*  eqlist = (uint32_t*)(smem + 67584);
  uint32_t*  shv    = (uint32_t*)(smem + 71680);     // [0]=pref [1]=kneed [2]=eqcnt [3]=selcnt

  const int row = blockIdx.x;
  const int tid = threadIdx.x;
  const float tau = tau_p[0];
  const int k = k_p[0];

  // uniform 64-bit row base addresses -> SGPR pairs for GVS-mode async loads
  const uint64_t rowS = (uint64_t)(uintptr_t)(scores + (size_t)row * N_NODES);
  const uint64_t rowN = (uint64_t)(uintptr_t)(noise  + (size_t)row * N_NODES);

  // LDS byte offsets (generic pointer low 32 bits == as(3) offset)
  const uint32_t ldsS = (uint32_t)(uintptr_t)bufS;
  const uint32_t ldsN = (uint32_t)(uintptr_t)bufN;

  // ---- CDNA5 async copy: global -> LDS, 16B per lane per instruction ----
  #pragma unroll
  for (int j = 0; j < VEC4_PER_THREAD; ++j) {
    uint32_t byteOff = (uint32_t)((j * BLOCK + tid) * 16);
    uint32_t la = ldsS + byteOff;
    uint32_t lb = ldsN + byteOff;
    asm volatile("global_load_async_to_lds_b128 %0, %1, %2"
                 :: "v"(la), "v"(byteOff), "s"(rowS) : "memory");
    asm volatile("global_load_async_to_lds_b128 %0, %1, %2"
                 :: "v"(lb), "v"(byteOff), "s"(rowN) : "memory");
  }
  asm volatile("s_wait_asynccnt 0" ::: "memory");
  __syncthreads();

  // ---- build sortable keys in place (overwrite bufS) ----
  #pragma unroll
  for (int j = 0; j < VEC4_PER_THREAD; ++j) {
    int g4 = j * BLOCK + tid;
    float4 s4 = ((const float4*)bufS)[g4];
    float4 n4 = ((const float4*)bufN)[g4];
    uint4 kk;
    kk.x = key_of(s4.x / tau + n4.x);
    kk.y = key_of(s4.y / tau + n4.y);
    kk.z = key_of(s4.z / tau + n4.z);
    kk.w = key_of(s4.w / tau + n4.w);
    ((uint4*)keys)[g4] = kk;
  }
  if (tid == 0) { shv[0] = 0u; shv[1] = (uint32_t)k; }
  __syncthreads();

  // ---- MSB-first radix select of the k-th largest key ----
  for (int pass = 0; pass < 4; ++pass) {
    const int shift = 24 - pass * 8;
    hist[tid] = 0u;            // BLOCK == 256 bins
    __syncthreads();
    const uint32_t prefBits = shv[0];
    const uint32_t maskHigh = (uint32_t)(0xFFFFFFFFull << (shift + 8));
    #pragma unroll
    for (int j = 0; j < VEC4_PER_THREAD; ++j) {
      uint4 kk = ((const uint4*)keys)[j * BLOCK + tid];
      if ((kk.x & maskHigh) == prefBits) atomicAdd(&hist[(kk.x >> shift) & 255u], 1u);
      if ((kk.y & maskHigh) == prefBits) atomicAdd(&hist[(kk.y >> shift) & 255u], 1u);
      if ((kk.z & maskHigh) == prefBits) atomicAdd(&hist[(kk.z >> shift) & 255u], 1u);
      if ((kk.w & maskHigh) == prefBits) atomicAdd(&hist[(kk.w >> shift) & 255u], 1u);
    }
    __syncthreads();

    // parallel inclusive suffix sum over 256 bins (Hillis-Steele, 8 steps):
    // scan[d] = sum(hist[d..255])
    scan[tid] = hist[tid];
    __syncthreads();
    #pragma unroll
    for (int off = 1; off < 256; off <<= 1) {
      uint32_t v = scan[tid];
      uint32_t add = (tid + off < 256) ? scan[tid + off] : 0u;
      __syncthreads();
      scan[tid] = v + add;
      __syncthreads();
    }

    const uint32_t kn = shv[1];
    const uint32_t sufIncl = scan[tid];
    const uint32_t sufExcl = (tid < 255) ? scan[tid + 1] : 0u;
    __syncthreads();
    if (sufExcl < kn && sufIncl >= kn) {   // exactly one thread wins
      shv[0] = prefBits | ((uint32_t)tid << shift);
      shv[1] = kn - sufExcl;               // how many still needed inside this bucket
    }
    __syncthreads();
  }
  const uint32_t T = shv[0];        // exact k-th largest key
  const uint32_t needEq = shv[1];   // ties at T to include (>=1)

  // ---- collect tie indices; pick the `needEq` smallest (top_k stability) ----
  if (tid == 0) shv[2] = 0u;
  __syncthreads();
  #pragma unroll
  for (int j = 0; j < VEC4_PER_THREAD; ++j) {
    int g4 = j * BLOCK + tid;
    uint4 kk = ((const uint4*)keys)[g4];
    uint32_t base = (uint32_t)g4 * 4u;
    if (kk.x == T) { uint32_t p = atomicAdd(&shv[2], 1u); if (p < 1024u) eqlist[p] = base + 0u; }
    if (kk.y == T) { uint32_t p = atomicAdd(&shv[2], 1u); if (p < 1024u) eqlist[p] = base + 1u; }
    if (kk.z == T) { uint32_t p = atomicAdd(&shv[2], 1u); if (p < 1024u) eqlist[p] = base + 2u; }
    if (kk.w == T) { uint32_t p = atomicAdd(&shv[2], 1u); if (p < 1024u) eqlist[p] = base + 3u; }
  }
  __syncthreads();
  if (tid == 0) {
    uint32_t cnt = shv[2] < 1024u ? shv[2] : 1024u;
    uint32_t need = needEq < cnt ? needEq : cnt;
    for (uint32_t i = 0; i < need; ++i) {       // partial selection sort: smallest indices first
      uint32_t mi = i;
      for (uint32_t j2 = i + 1; j2 < cnt; ++j2)
        if (eqlist[j2] < eqlist[mi]) mi = j2;
      uint32_t t = eqlist[i]; eqlist[i] = eqlist[mi]; eqlist[mi] = t;
    }
    shv[3] = need;
  }
  __syncthreads();
  const uint32_t selCnt = shv[3];

  // ---- write k-hot mask, coalesced float4 stores ----
  float* __restrict__ rowOut = out + (size_t)row * N_NODES;
  #pragma unroll
  for (int j = 0; j < VEC4_PER_THREAD; ++j) {
    int g4 = j * BLOCK + tid;
    uint4 kk = ((const uint4*)keys)[g4];
    uint32_t base = (uint32_t)g4 * 4u;
    float4 v;
    uint32_t kv[4] = {kk.x, kk.y, kk.z, kk.w};
    float    fv[4];
    #pragma unroll
    for (int c = 0; c < 4; ++c) {
      float r = 0.0f;
      if (kv[c] > T) r = 1.0f;
      else if (kv[c] == T) {
        for (uint32_t i = 0; i < selCnt; ++i)
          if (eqlist[i] == base + (uint32_t)c) { r = 1.0f; break; }
      }
      fv[c] = r;
    }
    v.x = fv[0]; v.y = fv[1]; v.z = fv[2]; v.w = fv[3];
    ((float4*)rowOut)[g4] = v;
  }
}

extern "C" void kernel_launch(void* const* d_in, const int* in_sizes, int n_in,
                              void* d_out, int out_size, void* d_ws, size_t ws_size,
                              hipStream_t stream) {
  const float* scores = (const float*)d_in[0];
  const float* noise  = (const float*)d_in[1];
  const float* tau    = (const float*)d_in[2];
  const int*   k      = (const int*)d_in[3];
  float* out = (float*)d_out;
  (void)in_sizes; (void)n_in; (void)out_size; (void)d_ws; (void)ws_size;

  hipLaunchKernelGGL(topk_khot_kernel, dim3(N_NODES), dim3(BLOCK), SMEM_BYTES, stream,
                     scores, noise, tau, k, out);
}